// VectorQuant_81114752352324
// MI455X (gfx1250) — compile-verified
//
#include <hip/hip_runtime.h>
#include <math.h>

// Problem constants: N=32, S=4096, C=1, K=512, V=128
#define NS     131072           // N*S*C rows
#define VDIM   128
#define KCODES 512
static constexpr float TN = 0.67882250993908571f;  // 0.06 * sqrt(128)

typedef __attribute__((ext_vector_type(2))) float v2f;
typedef __attribute__((ext_vector_type(8))) float v8f;

// ---- workspace layout (float units) ----
#define WS_EMBN 0                       // 512*128 normalized codes, row-major
#define WS_E2   (KCODES * VDIM)         // 512 per-code squared norms
#define WS_HIST (WS_E2 + KCODES)        // 512 uint32 histogram bins

// ---- output layout (float units, concatenated in return order) ----
#define OUT0_OFF 0                      // (N,S,C,V) = 16777216 floats
#define OUT1_OFF (NS * VDIM)
#define OUT2_OFF (OUT1_OFF + NS)
#define ENT_OFF  (OUT2_OFF + NS)

// ------------------------------------------------------------------
// Kernel 0: zero the histogram (ws is poisoned; must be deterministic)
// ------------------------------------------------------------------
__global__ void vq_zero_hist(float* ws) {
  unsigned* hist = reinterpret_cast<unsigned*>(ws + WS_HIST);
  hist[threadIdx.x] = 0u;   // <<<1, 512>>>
}

// ------------------------------------------------------------------
// Kernel 1: normalize codebook, store embN (row-major) + e2
// one wave per code row; <<<64, 256>>>
// ------------------------------------------------------------------
__global__ void vq_prep_emb(const float* __restrict__ emb0, float* __restrict__ ws) {
  const int wave = (blockIdx.x * blockDim.x + threadIdx.x) >> 5;
  const int lane = threadIdx.x & 31;
  if (wave >= KCODES) return;

  const float4 v = reinterpret_cast<const float4*>(emb0 + wave * VDIM)[lane];
  float s = v.x * v.x + v.y * v.y + v.z * v.z + v.w * v.w;
  for (int m = 16; m; m >>= 1) s += __shfl_xor(s, m, 32);

  const float scale = TN / sqrtf(s);
  float4 o;
  o.x = v.x * scale; o.y = v.y * scale; o.z = v.z * scale; o.w = v.w * scale;
  reinterpret_cast<float4*>(ws + WS_EMBN + wave * VDIM)[lane] = o;

  float e2p = o.x * o.x + o.y * o.y + o.z * o.z + o.w * o.w;
  for (int m = 16; m; m >>= 1) e2p += __shfl_xor(e2p, m, 32);
  if (lane == 0) ws[WS_E2 + wave] = e2p;
}

// ------------------------------------------------------------------
// Kernel 2: main VQ kernel. <<<2048, 128>>> : 4 waves/block, 64 rows/block,
// each wave owns a 16-row M-tile. FP32 WMMA (16x16x4) over K=128, all 512 codes.
// ------------------------------------------------------------------
__global__ void __launch_bounds__(128) vq_main(const float* __restrict__ x0,
                                               float* __restrict__ ws,
                                               float* __restrict__ out) {
  __shared__ float xs[64][132];   // normalized x tile, padded stride (33792 B)
  __shared__ float ec[32][132];   // codebook chunk [code_local][feat] (16896 B)
  __shared__ float ec2[32];       // e2 of chunk codes
  __shared__ float xsum[64];      // ||x0_row||^2
  __shared__ float xscale[64];    // tn / ||x0_row||

  const int tid  = threadIdx.x;
  const int lane = tid & 31;
  const int wv   = tid >> 5;          // wave id 0..3
  const int wrow = wv * 16;           // this wave's row base in LDS tile
  const int m    = lane & 15;         // matrix row (A) / column (B,C) within tile
  const int half = lane >> 4;         // K-group selector (A/B), row-group (C)
  const int rowg0 = blockIdx.x * 64;  // global row base of block

  // ---- Phase 1: load + normalize this wave's 16 x-rows into LDS ----
  for (int r = 0; r < 16; ++r) {
    const int row  = wrow + r;
    const float4 v = reinterpret_cast<const float4*>(x0 + (size_t)(rowg0 + row) * VDIM)[lane];
    float s = v.x * v.x + v.y * v.y + v.z * v.z + v.w * v.w;
    for (int mk = 16; mk; mk >>= 1) s += __shfl_xor(s, mk, 32);
    const float scale = TN / sqrtf(s);
    float4 o;
    o.x = v.x * scale; o.y = v.y * scale; o.z = v.z * scale; o.w = v.w * scale;
    *reinterpret_cast<float4*>(&xs[row][lane * 4]) = o;   // 16B-aligned (132*4=528)
    if (lane == 0) { xsum[row] = s; xscale[row] = scale; }
  }

  // ---- Phase 2: stream codebook in 32-code chunks, WMMA + running argmin ----
  float bestd[8];
  int   bestk[8];
#pragma unroll
  for (int v = 0; v < 8; ++v) { bestd[v] = 3.4e38f; bestk[v] = 0; }

  for (int ch = 0; ch < KCODES / 32; ++ch) {
    __syncthreads();   // previous chunk fully consumed before overwrite
    // stage chunk: 32 codes x 128 feats = 1024 float4, 8 per thread (coalesced)
#pragma unroll
    for (int i = 0; i < 8; ++i) {
      const int f4   = i * 128 + tid;     // 0..1023
      const int code = f4 >> 5;           // 0..31
      const int c4   = f4 & 31;           // float4 index within row
      const float4 t = *reinterpret_cast<const float4*>(
          ws + WS_EMBN + (size_t)(ch * 32 + code) * VDIM + c4 * 4);
      *reinterpret_cast<float4*>(&ec[code][c4 * 4]) = t;
    }
    if (tid < 32) ec2[tid] = ws[WS_E2 + ch * 32 + tid];
    __syncthreads();

    // two 16-code tiles per chunk
#pragma unroll
    for (int t = 0; t < 2; ++t) {
      const int nloc = t * 16 + m;        // this lane's code column (local)
      v8f acc = {};
#pragma unroll
      for (int kk = 0; kk < 32; ++kk) {   // K = 128 / 4
        const int k0 = kk * 4 + half * 2;
        v2f a, b;
        a.x = xs[wrow + m][k0];  a.y = xs[wrow + m][k0 + 1];   // A 16x4 layout
        b.x = ec[nloc][k0];      b.y = ec[nloc][k0 + 1];       // B 4x16 layout
        acc = __builtin_amdgcn_wmma_f32_16x16x4_f32(
            false, a, false, b, (short)0, acc, false, false);
      }
      const float e2v  = ec2[nloc];
      const int   kglb = ch * 32 + t * 16 + m;
#pragma unroll
      for (int v = 0; v < 8; ++v) {       // lane holds rows v + 8*half, col nloc
        const float d = e2v - 2.0f * acc[v];
        if (d < bestd[v]) { bestd[v] = d; bestk[v] = kglb; }
      }
    }
  }

  // ---- Phase 3: cross-lane argmin within each 16-lane half (tie -> lower k) ----
#pragma unroll
  for (int v = 0; v < 8; ++v) {
    float d = bestd[v];
    int   k = bestk[v];
    for (int msk = 8; msk; msk >>= 1) {
      const float od = __shfl_xor(d, msk, 32);
      const int   ok = __shfl_xor(k, msk, 32);
      if (od < d || (od == d && ok < k)) { d = od; k = ok; }
    }
    bestd[v] = d; bestk[v] = k;
  }

  // ---- Phase 4: scalar outputs (out1/out2) + histogram ----
  unsigned* hist = reinterpret_cast<unsigned*>(ws + WS_HIST);
  if (m == 0) {          // lanes 0 and 16: one per row-group
#pragma unroll
    for (int v = 0; v < 8; ++v) {
      const int row  = wrow + v + 8 * half;
      const int grow = rowg0 + row;
      const float sc = xscale[row];
      const float x2 = sc * sc * xsum[row];
      const float o1 = x2 + bestd[v];                  // ||x - q||^2
      const float dd = (sc - 1.0f) * (sc - 1.0f) * xsum[row];  // ||x - x0||^2
      out[OUT1_OFF + grow] = o1;
      out[OUT2_OFF + grow] = o1 + dd;
      atomicAdd(hist + bestk[v], 1u);
    }
  }

  // ---- Phase 5: out0 = gather winning code rows (coalesced float4) ----
#pragma unroll
  for (int v = 0; v < 8; ++v) {
    const int kA = __shfl(bestk[v], 0, 32);    // result for row wrow+v   (half 0)
    const int kB = __shfl(bestk[v], 16, 32);   // result for row wrow+v+8 (half 1)
    const size_t rowA = (size_t)(rowg0 + wrow + v);
    const size_t rowB = rowA + 8;
    const float4 eA = reinterpret_cast<const float4*>(ws + WS_EMBN + (size_t)kA * VDIM)[lane];
    reinterpret_cast<float4*>(out + OUT0_OFF)[rowA * 32 + lane] = eA;
    const float4 eB = reinterpret_cast<const float4*>(ws + WS_EMBN + (size_t)kB * VDIM)[lane];
    reinterpret_cast<float4*>(out + OUT0_OFF)[rowB * 32 + lane] = eB;
  }
}

// ------------------------------------------------------------------
// Kernel 3: entropy of code usage histogram. <<<1, 512>>>
// ------------------------------------------------------------------
__global__ void vq_entropy(const float* __restrict__ ws, float* __restrict__ out) {
  __shared__ float red[512];
  const unsigned* hist = reinterpret_cast<const unsigned*>(ws + WS_HIST);
  const unsigned h = hist[threadIdx.x];
  const float p = (float)h / (float)NS;
  red[threadIdx.x] = (h > 0u) ? (-p * logf(p)) : 0.0f;
  __syncthreads();
  for (int s = 256; s; s >>= 1) {
    if (threadIdx.x < s) red[threadIdx.x] += red[threadIdx.x + s];
    __syncthreads();
  }
  if (threadIdx.x == 0) out[ENT_OFF] = red[0];
}

// ------------------------------------------------------------------
extern "C" void kernel_launch(void* const* d_in, const int* in_sizes, int n_in,
                              void* d_out, int out_size, void* d_ws, size_t ws_size,
                              hipStream_t stream) {
  const float* x0   = (const float*)d_in[0];   // (32,4096,1,128) f32
  const float* emb0 = (const float*)d_in[1];   // (1,512,128) f32
  float* out = (float*)d_out;
  float* ws  = (float*)d_ws;
  (void)in_sizes; (void)n_in; (void)out_size; (void)ws_size;

  vq_zero_hist<<<1, 512, 0, stream>>>(ws);
  vq_prep_emb<<<64, 256, 0, stream>>>(emb0, ws);
  vq_main<<<NS / 64, 128, 0, stream>>>(x0, ws, out);
  vq_entropy<<<1, 512, 0, stream>>>(ws, out);
}